// HisCoM_4200478015919
// MI455X (gfx1250) — compile-verified
//
#include <hip/hip_runtime.h>
#include <cstdint>
#include <cstddef>

// Problem constants (match reference)
#define B_ 2048
#define G_ 20000
#define P_ 2000
#define E_ 100000
#define BN_EPS_ 1e-5f

typedef __attribute__((ext_vector_type(2))) float v2f;
typedef __attribute__((ext_vector_type(8))) float v8f;

// ---------------------------------------------------------------------------
// Kernel 1: g[b,g] = dot3(x[b,g,:], w_gene), stored TRANSPOSED as g_t[G,B].
// 64(b) x 32(g) LDS tile: x reads coalesced along g, g_t writes coalesced
// along b. This makes the downstream pathway gather coalesced across batch.
// grid = (G/32, B/64) = (625, 32), block = (32, 8)
// ---------------------------------------------------------------------------
__global__ __launch_bounds__(256) void k_gene_enc_t(const float* __restrict__ x,
                                                    const float* __restrict__ wg,
                                                    float* __restrict__ g_t) {
    __shared__ float tile[64][33];           // [b_local][g_local], padded
    const int tx = threadIdx.x;              // 0..31  (gene within tile)
    const int ty = threadIdx.y;              // 0..7
    const int g0 = blockIdx.x * 32;          // gene base
    const int b0 = blockIdx.y * 64;          // batch base
    const float w0 = wg[0], w1 = wg[1], w2 = wg[2];

#pragma unroll
    for (int s = 0; s < 8; ++s) {
        const int bl = ty * 8 + s;                                  // 0..63
        const size_t base = ((size_t)(b0 + bl) * G_ + (g0 + tx)) * 3;
        tile[bl][tx] = fmaf(x[base], w0, fmaf(x[base + 1], w1, x[base + 2] * w2));
    }
    __syncthreads();
#pragma unroll
    for (int k = 0; k < 4; ++k) {
        const int gl = ty * 4 + k;                                  // 0..31
        const size_t row = (size_t)(g0 + gl) * B_ + b0;
        g_t[row + tx]      = tile[tx][gl];
        g_t[row + tx + 32] = tile[tx + 32][gl];
    }
}

// ---------------------------------------------------------------------------
// Kernel 2: CSR offsets for sorted seg_id: off[p] = lower_bound(seg_id, p),
// off[P] = E.
// ---------------------------------------------------------------------------
__global__ void k_seg_off(const int* __restrict__ seg, int* __restrict__ off) {
    const int p = blockIdx.x * blockDim.x + threadIdx.x;
    if (p > P_) return;
    if (p == P_) { off[P_] = E_; return; }
    int lo = 0, hi = E_;
    while (lo < hi) {
        const int mid = (lo + hi) >> 1;
        if (seg[mid] < p) lo = mid + 1; else hi = mid;
    }
    off[p] = lo;
}

// ---------------------------------------------------------------------------
// Kernel 3: Pmat[p,b] = sum_{e in segment p} g_t[gene_idx[e], b] * w_path[e]
// fused with BatchNorm statistics (mean / rsqrt(var+eps) over batch).
// One block per pathway; e is block-uniform -> scalar loads; g_t reads are
// coalesced (256 consecutive b) and L2-resident.
// grid = P, block = 256
// ---------------------------------------------------------------------------
__global__ __launch_bounds__(256) void k_pathway(const float* __restrict__ g_t,
                                                 const int* __restrict__ gene_idx,
                                                 const float* __restrict__ w_path,
                                                 const int* __restrict__ off,
                                                 float* __restrict__ pmat,
                                                 float* __restrict__ mean,
                                                 float* __restrict__ rstd) {
    const int p   = blockIdx.x;
    const int tid = threadIdx.x;
    const int s = off[p], e = off[p + 1];

    float lsum = 0.f, lsq = 0.f;
    for (int b0 = 0; b0 < B_; b0 += 256) {
        const int b = b0 + tid;
        float acc = 0.f;
        for (int i = s; i < e; ++i) {
            const int gi = gene_idx[i];
            acc = fmaf(g_t[(size_t)gi * B_ + b], w_path[i], acc);
        }
        pmat[(size_t)p * B_ + b] = acc;
        lsum += acc;
        lsq = fmaf(acc, acc, lsq);
    }

    __shared__ float rs[256], rq[256];
    rs[tid] = lsum; rq[tid] = lsq;
    __syncthreads();
    for (int st = 128; st > 0; st >>= 1) {
        if (tid < st) { rs[tid] += rs[tid + st]; rq[tid] += rq[tid + st]; }
        __syncthreads();
    }
    if (tid == 0) {
        const float m = rs[0] * (1.f / (float)B_);
        const float v = rq[0] * (1.f / (float)B_) - m * m;
        mean[p] = m;
        rstd[p] = rsqrtf(v + BN_EPS_);
    }
}

// ---------------------------------------------------------------------------
// Kernel 4: out[b] = sigmoid( sum_p z[p,b]*w_fc[p] + b_fc ) with
// z[p,b] = (Pmat[p,b]-mean[p])*rstd[p]*gamma[p] + beta[p].
// The GEMV is done with V_WMMA_F32_16X16X4_F32: A = 16(batch) x 4(path) tile
// of z (A layout: lanes 0-15 -> K=0/1 in v[0]/v[1], lanes 16-31 -> K=2/3),
// B[k][n] = w_fc[k] broadcast to all 16 columns, C accumulated over P/4=500
// steps. D rows are constant -> lanes 0 and 16 extract the 16 results.
// grid = 16, block = 256 (8 waves x 16 batch rows = 2048 = B exactly,
// so EXEC is all ones everywhere the wmma executes).
// ---------------------------------------------------------------------------
__global__ __launch_bounds__(256) void k_fc_wmma(const float* __restrict__ pmat,
                                                 const float* __restrict__ mean,
                                                 const float* __restrict__ rstd,
                                                 const float* __restrict__ gamma,
                                                 const float* __restrict__ beta,
                                                 const float* __restrict__ w_fc,
                                                 const float* __restrict__ b_fc,
                                                 float* __restrict__ out) {
    const int tid   = threadIdx.x;
    const int wave  = tid >> 5;
    const int lane  = tid & 31;
    const int m     = lane & 15;
    const int half  = lane >> 4;                 // 0: K=0,1   1: K=2,3
    const int baseB = blockIdx.x * 128 + wave * 16;
    const int b     = baseB + m;

    v8f c = {};
    for (int p0 = 0; p0 < P_; p0 += 4) {
        const int pa = p0 + half * 2;
        const int pb = pa + 1;
        const float z0 = (pmat[(size_t)pa * B_ + b] - mean[pa]) * rstd[pa] * gamma[pa] + beta[pa];
        const float z1 = (pmat[(size_t)pb * B_ + b] - mean[pb]) * rstd[pb] * gamma[pb] + beta[pb];
        v2f a;  a.x  = z0;        a.y  = z1;
        v2f bb; bb.x = w_fc[pa];  bb.y = w_fc[pb];
        // 8 args: (neg_a, A, neg_b, B, c_mod, C, reuse_a, reuse_b)
        c = __builtin_amdgcn_wmma_f32_16x16x4_f32(false, a, false, bb,
                                                  (short)0, c, false, false);
    }

    const float bias = b_fc[0];
    if (lane == 0) {
#pragma unroll
        for (int v = 0; v < 8; ++v) {            // D rows 0..7
            const float t = c[v] + bias;
            out[baseB + v] = 1.f / (1.f + __expf(-t));
        }
    } else if (lane == 16) {
#pragma unroll
        for (int v = 0; v < 8; ++v) {            // D rows 8..15
            const float t = c[v] + bias;
            out[baseB + 8 + v] = 1.f / (1.f + __expf(-t));
        }
    }
}

// ---------------------------------------------------------------------------
// Host-side launcher
// ---------------------------------------------------------------------------
extern "C" void kernel_launch(void* const* d_in, const int* in_sizes, int n_in,
                              void* d_out, int out_size, void* d_ws, size_t ws_size,
                              hipStream_t stream) {
    (void)in_sizes; (void)n_in; (void)out_size; (void)ws_size;

    const float* x        = (const float*)d_in[0];   // [B,G,3]
    const float* w_gene   = (const float*)d_in[1];   // [3]
    const float* w_path   = (const float*)d_in[2];   // [E]
    const int*   gene_idx = (const int*)  d_in[3];   // [E]
    const int*   seg_id   = (const int*)  d_in[4];   // [E] sorted
    const float* gamma    = (const float*)d_in[5];   // [P]
    const float* beta     = (const float*)d_in[6];   // [P]
    const float* w_fc     = (const float*)d_in[7];   // [P]
    const float* b_fc     = (const float*)d_in[8];   // [1]
    float*       out      = (float*)d_out;           // [B,1]

    // Workspace layout (256B aligned):
    //   g_t  : [G,B] floats   = 163,840,000 B
    //   pmat : [P,B] floats   =  16,384,000 B
    //   mean : [P] floats
    //   rstd : [P] floats
    //   off  : [P+1] ints
    char* ws = (char*)d_ws;
    float* g_t  = (float*)(ws);
    float* pmat = (float*)(ws + 163840000);
    float* mean = (float*)(ws + 180224000);
    float* rstd = (float*)(ws + 180232192);
    int*   off  = (int*)  (ws + 180240384);

    k_gene_enc_t<<<dim3(G_ / 32, B_ / 64), dim3(32, 8), 0, stream>>>(x, w_gene, g_t);
    k_seg_off<<<(P_ + 1 + 255) / 256, 256, 0, stream>>>(seg_id, off);
    k_pathway<<<P_, 256, 0, stream>>>(g_t, gene_idx, w_path, off, pmat, mean, rstd);
    k_fc_wmma<<<B_ / 128, 256, 0, stream>>>(pmat, mean, rstd, gamma, beta, w_fc, b_fc, out);
}